// ScaleDotProductAttention_53884659696358
// MI455X (gfx1250) — compile-verified
//
#include <hip/hip_runtime.h>
#include <math.h>

typedef __attribute__((ext_vector_type(2))) float v2f;
typedef __attribute__((ext_vector_type(4))) float v4f;
typedef __attribute__((ext_vector_type(8))) float v8f;

#define BATCH   16
#define LQ      2048
#define LK      2048
#define DIM     128
#define QTILE   16
#define NWAVES  8
#define LDS_STRIDE 2052           // 2048 + 4 pad -> bank-conflict-free A reads
#define NEG_INF_VAL (-1.0e9f)

// dynamic LDS layout (floats):
//   s_S      : 16 * 2052          (score strip -> probabilities)
//   s_red    : 256                (per-row partial reductions)
//   s_rowmax : 16
//   s_rowrcp : 16
#define SMEM_FLOATS (QTILE * LDS_STRIDE + 256 + 16 + 16)

__launch_bounds__(256)
__global__ void sdpa_f32wmma_kernel(const float* __restrict__ Q,
                                    const float* __restrict__ K,
                                    const float* __restrict__ V,
                                    const unsigned char* __restrict__ mask,
                                    float* __restrict__ out_o,   // [B, Lq, D]
                                    float* __restrict__ out_p) { // [B, Lq, Lk]
    extern __shared__ float smem[];
    float* s_S      = smem;
    float* s_red    = smem + QTILE * LDS_STRIDE;
    float* s_rowmax = s_red + 256;
    float* s_rowrcp = s_rowmax + 16;

    const int blk  = blockIdx.x;        // 0 .. B*Lq/16 - 1 = 2047
    const int b    = blk >> 7;          // / (Lq/16)
    const int q0   = (blk & 127) << 4;  // query-row base

    const int tid   = threadIdx.x;
    const int lane  = tid & 31;
    const int wave  = tid >> 5;
    const int lrow  = lane & 15;        // M (or N) index within tile
    const int khalf = lane >> 4;        // selects k-pair {0,1} vs {2,3}

    const float scale = 0.08838834764831845f; // 1/sqrt(128)

    // ---------------- load Q block as 32 A-fragments (K=128 in steps of 4) ---
    v2f a[32];
    {
        const float* qrow = Q + ((size_t)b * LQ + (q0 + lrow)) * DIM + 2 * khalf;
        #pragma unroll
        for (int ks = 0; ks < 32; ++ks)
            a[ks] = *(const v2f*)(qrow + ks * 4);
    }

    // ---------------- Phase 1: S = mask(scale * Q K^T) into LDS -------------
    for (int t = 0; t < 16; ++t) {
        const int kt0 = ((t * NWAVES + wave) << 4);   // this wave's key-tile base
        v8f c = {};
        const float* krow = K + ((size_t)b * LK + (kt0 + lrow)) * DIM + 2 * khalf;
        #pragma unroll
        for (int ks = 0; ks < 32; ++ks) {
            v2f bf = *(const v2f*)(krow + ks * 4);
            c = __builtin_amdgcn_wmma_f32_16x16x4_f32(
                    false, a[ks], false, bf, (short)0, c, false, false);
        }
        const unsigned char* mbase =
            mask + ((size_t)b * LQ + q0) * LK + kt0;
        #pragma unroll
        for (int r = 0; r < 8; ++r) {
            const int m = r + 8 * khalf;
            const int n = lrow;
            float s = c[r] * scale;
            if (mbase[(size_t)m * LK + n]) s = NEG_INF_VAL;
            s_S[m * LDS_STRIDE + kt0 + n] = s;
        }
    }
    __syncthreads();

    // ---------------- Phase 2: softmax rows in LDS ---------------------------
    {
        const int row = tid >> 4;      // 0..15
        const int sub = tid & 15;      // 16 threads per row
        float* rowp = s_S + row * LDS_STRIDE;

        float mx = -3.0e38f;
        for (int ci = sub; ci < LK; ci += 16)
            mx = fmaxf(mx, rowp[ci]);
        s_red[row * 16 + sub] = mx;
        __syncthreads();
        if (sub == 0) {
            float m2 = s_red[row * 16];
            #pragma unroll
            for (int i = 1; i < 16; ++i) m2 = fmaxf(m2, s_red[row * 16 + i]);
            s_rowmax[row] = m2;
        }
        __syncthreads();

        const float rm = s_rowmax[row];
        float partial = 0.0f;
        for (int ci = sub; ci < LK; ci += 16) {
            float e = expf(rowp[ci] - rm);
            rowp[ci] = e;
            partial += e;
        }
        __syncthreads();
        s_red[row * 16 + sub] = partial;
        __syncthreads();
        if (sub == 0) {
            float sum = 0.0f;
            #pragma unroll
            for (int i = 0; i < 16; ++i) sum += s_red[row * 16 + i];
            s_rowrcp[row] = 1.0f / sum;
        }
        __syncthreads();

        const float rcp = s_rowrcp[row];
        for (int ci = sub; ci < LK; ci += 16)
            rowp[ci] *= rcp;
    }
    __syncthreads();

    // ---------------- write attention weights (coalesced float4) ------------
    {
        float* pbase = out_p + ((size_t)b * LQ + q0) * LK;
        #pragma unroll
        for (int row = 0; row < 16; ++row) {
            #pragma unroll
            for (int ch = 0; ch < 2; ++ch) {
                const int col = (ch << 10) + (tid << 2);   // 256 thr * 4 floats * 2
                *(v4f*)(pbase + (size_t)row * LK + col) =
                    *(const v4f*)(s_S + row * LDS_STRIDE + col);
            }
        }
    }

    // ---------------- Phase 3: O = P @ V (each wave owns 16 cols of D) ------
    {
        const int d0 = wave << 4;
        v8f o = {};
        #pragma unroll 4
        for (int ks = 0; ks < 512; ++ks) {
            const int kb = ks * 4 + 2 * khalf;
            v2f ap = *(const v2f*)(s_S + lrow * LDS_STRIDE + kb); // A: P[lrow][kb..kb+1]
            const float* vp = V + ((size_t)b * LK + kb) * DIM + d0 + lrow;
            v2f bf;
            bf.x = vp[0];        // V[kb  ][d0+lrow]
            bf.y = vp[DIM];      // V[kb+1][d0+lrow]
            o = __builtin_amdgcn_wmma_f32_16x16x4_f32(
                    false, ap, false, bf, (short)0, o, false, false);
        }
        float* obase = out_o + ((size_t)b * LQ + q0) * DIM + d0;
        #pragma unroll
        for (int r = 0; r < 8; ++r) {
            const int m = r + 8 * khalf;
            obase[(size_t)m * DIM + lrow] = o[r];
        }
    }
}

extern "C" void kernel_launch(void* const* d_in, const int* in_sizes, int n_in,
                              void* d_out, int out_size, void* d_ws, size_t ws_size,
                              hipStream_t stream) {
    const float*         q    = (const float*)d_in[0];
    const float*         k    = (const float*)d_in[1];
    const float*         v    = (const float*)d_in[2];
    const unsigned char* mask = (const unsigned char*)d_in[3]; // jnp.bool_ = 1 byte

    float* out_o = (float*)d_out;                                   // B*Lq*D
    float* out_p = (float*)d_out + (size_t)BATCH * LQ * DIM;        // B*Lq*Lk

    const dim3 grid(BATCH * (LQ / QTILE));   // 2048 blocks
    const dim3 block(256);                   // 8 wave32
    const size_t smem_bytes = SMEM_FLOATS * sizeof(float);          // ~132.5 KB

    hipLaunchKernelGGL(sdpa_f32wmma_kernel, grid, block, smem_bytes, stream,
                       q, k, v, mask, out_o, out_p);
}